// SparseModelWrapper_58222576664867
// MI455X (gfx1250) — compile-verified
//
#include <hip/hip_runtime.h>

// out[p, :] = x[p, :] @ W + b  for p in [0, P)
// (the reference's sort/inverse permutations cancel exactly; see analysis)
//
// WMMA mapping (V_WMMA_F32_16X16X4_F32, wave32):
//   A (16x4) : 16 points x [x0, x1, x2, 1.0]   -> 2 VGPRs/lane (v2f)
//   B (4x16) : [W row0; W row1; W row2; bias]  -> 2 VGPRs/lane (v2f), built once
//   C = 0 (bias folded into A's K=3 column times B's K=3 row)
//   D (16x16): column n (n<3) = output feature n of point (row m)

typedef __attribute__((ext_vector_type(2))) float v2f;
typedef __attribute__((ext_vector_type(8))) float v8f;

#define TILE 16

__global__ __launch_bounds__(256) void affine_wmma_kernel(
    const float* __restrict__ x, const float* __restrict__ W,
    const float* __restrict__ bias, float* __restrict__ out, int ntiles)
{
    const int lane   = threadIdx.x & 31;
    const int wave   = blockIdx.x * (blockDim.x >> 5) + (threadIdx.x >> 5);
    const int nwaves = gridDim.x * (blockDim.x >> 5);

    const int  n  = lane & 15;     // column / K-row selector within half-wave
    const bool hi = lane >= 16;

    // ---- Build B operand (4x16) once per wave ----
    // VGPR0: lanes0-15 = B[K=0][n] = W[0][n], lanes16-31 = B[K=2][n] = W[2][n]
    // VGPR1: lanes0-15 = B[K=1][n] = W[1][n], lanes16-31 = B[K=3][n] = bias[n]
    float b0 = 0.0f, b1 = 0.0f;
    if (n < 3) {
        b0 = hi ? W[2 * 3 + n] : W[0 * 3 + n];
        b1 = hi ? bias[n]      : W[1 * 3 + n];
    }
    v2f B; B.x = b0; B.y = b1;

    for (int t = wave; t < ntiles; t += nwaves) {
        const int pb = t * TILE;       // first point of this 16-point tile
        const int p  = pb + n;         // this lane's point (row M = n)

        // Prefetch next grid-stride chunk (speculative; dropped if OOB).
        if (t + nwaves < ntiles)
            __builtin_prefetch(x + (size_t)(p + nwaves * TILE) * 3, 0, 1);

        // ---- A operand (16x4) ----
        // VGPR0: lanes0-15 K=0 (x0), lanes16-31 K=2 (x2)
        // VGPR1: lanes0-15 K=1 (x1), lanes16-31 K=3 (1.0)
        const float a0 = x[p * 3 + (hi ? 2 : 0)];
        const float a1 = hi ? 1.0f : x[p * 3 + 1];
        v2f A; A.x = a0; A.y = a1;

        v8f C = {};  // bias already folded in via B's K=3 row
        v8f D = __builtin_amdgcn_wmma_f32_16x16x4_f32(
            /*neg_a=*/false, A, /*neg_b=*/false, B,
            /*c_mod=*/(short)0, C, /*reuse_a=*/false, /*reuse_b=*/false);

        // ---- Scatter D back: VGPR r, lanes0-15 -> M=r, lanes16-31 -> M=r+8,
        // column = n; only columns 0..2 are real outputs.
        const int mbase = pb + (hi ? 8 : 0);
        if (n < 3) {
            #pragma unroll
            for (int r = 0; r < 8; ++r)
                out[(mbase + r) * 3 + n] = D[r];
        }
        // reconvergence restores full EXEC before next iteration's WMMA
    }
}

// Scalar fallback for P % 16 leftover points (none for the reference shapes).
__global__ void affine_tail_kernel(const float* __restrict__ x,
                                   const float* __restrict__ W,
                                   const float* __restrict__ bias,
                                   float* __restrict__ out, int start, int P)
{
    int p = start + blockIdx.x * blockDim.x + threadIdx.x;
    if (p >= P) return;
    float x0 = x[p * 3 + 0], x1 = x[p * 3 + 1], x2 = x[p * 3 + 2];
    #pragma unroll
    for (int f = 0; f < 3; ++f)
        out[p * 3 + f] =
            fmaf(x0, W[0 * 3 + f], fmaf(x1, W[1 * 3 + f], fmaf(x2, W[2 * 3 + f], bias[f])));
}

extern "C" void kernel_launch(void* const* d_in, const int* in_sizes, int n_in,
                              void* d_out, int out_size, void* d_ws, size_t ws_size,
                              hipStream_t stream)
{
    const float* x    = (const float*)d_in[0];   // [B, N, 3] flat
    const float* W    = (const float*)d_in[1];   // [3, 3]
    const float* bias = (const float*)d_in[2];   // [3]
    float*       out  = (float*)d_out;           // [B, N, 3] flat

    const int P      = in_sizes[0] / 3;          // total points
    const int ntiles = P / TILE;                 // full 16-point WMMA tiles
    const int tail   = P - ntiles * TILE;

    if (ntiles > 0) {
        // 4096 blocks x 8 waves = 32768 waves; 16 tiles each at P = 8.4M.
        dim3 grid(4096), block(256);
        affine_wmma_kernel<<<grid, block, 0, stream>>>(x, W, bias, out, ntiles);
    }
    if (tail > 0) {
        affine_tail_kernel<<<1, 64, 0, stream>>>(x, W, bias, out, ntiles * TILE, P);
    }
}